// GATWithEdgeFeatures_53283364274719
// MI455X (gfx1250) — compile-verified
//
#include <hip/hip_runtime.h>
#include <hip/hip_bf16.h>

#define NN 50000
#define NE 800000
#define INF_ 64
#define HF 64
#define NH 4
#define EDF 16
#define OUTF 32
#define NG 512

typedef __attribute__((ext_vector_type(2))) float v2f;
typedef __attribute__((ext_vector_type(8))) float v8f;
typedef unsigned int v4u __attribute__((ext_vector_type(4)));
typedef int v4i __attribute__((ext_vector_type(4)));
typedef int v8i __attribute__((ext_vector_type(8)));

#if defined(__HIP_DEVICE_COMPILE__) && __has_builtin(__builtin_amdgcn_tensor_load_to_lds) && \
    __has_builtin(__builtin_amdgcn_s_wait_tensorcnt)
#define USE_TDM 1
#else
#define USE_TDM 0
#endif

// ---------------- utility ----------------
__device__ __forceinline__ unsigned flipf(float f) {
  unsigned u = __float_as_uint(f);
  return (u & 0x80000000u) ? ~u : (u | 0x80000000u);
}
__device__ __forceinline__ float unflipf(unsigned u) {
  unsigned v = (u & 0x80000000u) ? (u ^ 0x80000000u) : ~u;
  return __uint_as_float(v);
}

__global__ void zero_f32(float* p, size_t n) {
  size_t i = (size_t)blockIdx.x * blockDim.x + threadIdx.x;
  size_t st = (size_t)gridDim.x * blockDim.x;
  for (; i < n; i += st) p[i] = 0.0f;
}

#if USE_TDM
// Issue a TDM copy of `bytes/4` f32 elements (one contiguous row) from global B
// into LDS at `lds_off`. Must be executed by exactly one wave; EXEC is ignored.
__device__ __forceinline__ void tdm_load_b(const float* B, unsigned lds_off, unsigned nelem) {
  unsigned long long ga = (unsigned long long)(const void*)B;
  v4u g0;
  g0[0] = 1u;                                               // count=1, user mode
  g0[1] = lds_off;                                          // LDS byte address
  g0[2] = (unsigned)ga;                                     // global_addr[31:0]
  g0[3] = (unsigned)((ga >> 32) & 0x1FFFFFFu) | (2u << 30); // addr[56:32] | type=2
  v8i g1;
  g1[0] = 0x00020000;                                  // data_size=4B, no multicast
  g1[1] = (int)((nelem & 0xFFFFu) << 16);              // tensor_dim0[15:0]
  g1[2] = (int)(((nelem >> 16) & 0xFFFFu) | (1u << 16)); // dim0[31:16] | tensor_dim1=1
  g1[3] = (int)((nelem & 0xFFFFu) << 16);              // tile_dim0 = nelem
  g1[4] = 1;                                           // tile_dim1=1, tile_dim2=0
  g1[5] = (int)nelem;                                  // tensor_dim0_stride
  g1[6] = 0;
  g1[7] = 0;
  v4i z4 = {0, 0, 0, 0};
#if __clang_major__ >= 23
  v8i z8 = {0, 0, 0, 0, 0, 0, 0, 0};
  __builtin_amdgcn_tensor_load_to_lds(g0, g1, z4, z4, z8, 0);
#else
  __builtin_amdgcn_tensor_load_to_lds(g0, g1, z4, z4, 0);
#endif
  __builtin_amdgcn_s_wait_tensorcnt(0);
}
#endif

// ---------------- WMMA f32 GEMM ----------------
// C[M,Nc] = act(bn(A(+A2) @ B + bias)); A row-major [M,K], B row-major [K,Nc].
// 256-thread block = 8 waves; each wave owns one 16-row strip and all Nc columns
// (NT = Nc/16 accumulators). B panel staged in LDS once per block (TDM when
// available), so the K-loop issues NT v_wmma per step fed by ds_load pairs.
template <int NT, bool HAS_A2, bool HAS_BN, bool RELU, bool USE_LDS>
__global__ __launch_bounds__(256) void gemm_wmma(
    const float* __restrict__ A, const float* __restrict__ A2,
    const float* __restrict__ B, const float* __restrict__ bias,
    const float* __restrict__ bn_g, const float* __restrict__ bn_b,
    const float* __restrict__ bn_m, const float* __restrict__ bn_v,
    float* __restrict__ C, int M, int K)
{
  constexpr int Nc = NT * 16;
  extern __shared__ float ldsB[];
  const int tid = threadIdx.x;

  if constexpr (USE_LDS) {
#if USE_TDM
    if (tid < 32) {  // wave 0 drives the Tensor Data Mover
      tdm_load_b(B, (unsigned)(size_t)(void*)&ldsB[0], (unsigned)(K * Nc));
    }
#else
    for (int i = tid; i < K * Nc; i += 256) ldsB[i] = B[i];
#endif
    __syncthreads();
  }

  const int wave = tid >> 5;
  const int mt = blockIdx.x * 8 + wave;
  if (mt * 16 >= M) return;

  const int lane = tid & 31;
  const int hl = lane >> 4;   // which K-pair within the 4-step (ISA A/B layout)
  const int l15 = lane & 15;
  const int m0 = mt * 16;

  const float* arow = A + (size_t)(m0 + l15) * K + 2 * hl;
  const float* arow2 = HAS_A2 ? (A2 + (size_t)(m0 + l15) * K + 2 * hl) : nullptr;

  v8f acc[NT] = {};
  for (int k = 0; k < K; k += 4) {
    float2 av = *(const float2*)(arow + k);
    if constexpr (HAS_A2) {
      float2 a2v = *(const float2*)(arow2 + k);
      av.x += a2v.x; av.y += a2v.y;
    }
    v2f a; a.x = av.x; a.y = av.y;
#pragma unroll
    for (int t = 0; t < NT; ++t) {
      int n = t * 16 + l15;
      v2f b;
      if constexpr (USE_LDS) {
        b.x = ldsB[(k + 2 * hl) * Nc + n];
        b.y = ldsB[(k + 2 * hl + 1) * Nc + n];
      } else {
        b.x = B[(size_t)(k + 2 * hl) * Nc + n];
        b.y = B[(size_t)(k + 2 * hl + 1) * Nc + n];
      }
      acc[t] = __builtin_amdgcn_wmma_f32_16x16x4_f32(false, a, false, b, (short)0,
                                                     acc[t], false, false);
    }
  }

#pragma unroll
  for (int t = 0; t < NT; ++t) {
    int n = t * 16 + l15;
    float bi = bias[n];
    float g = 1.0f, bb = 0.0f, mm = 0.0f, inv = 1.0f;
    if constexpr (HAS_BN) {
      g = bn_g[n]; bb = bn_b[n]; mm = bn_m[n]; inv = rsqrtf(bn_v[n] + 1e-5f);
    }
#pragma unroll
    for (int r = 0; r < 8; ++r) {
      int m = m0 + r + hl * 8;
      float v = acc[t][r] + bi;
      if constexpr (HAS_BN) v = (v - mm) * inv * g + bb;
      if constexpr (RELU) v = fmaxf(v, 0.0f);
      C[(size_t)m * Nc + n] = v;
    }
  }
}

// ---------------- fold attention vectors into small matrices ----------------
__global__ void precompute(const float* __restrict__ Wgat, const float* __restrict__ att_src,
                           const float* __restrict__ att_dst, const float* __restrict__ Wge,
                           const float* __restrict__ att_edge,
                           float* Asrc, float* Adst, float* WeA, float* Wstack)
{
  int t = threadIdx.x;  // 256 threads, one block
  int k = t >> 2, h = t & 3;
  float s1 = 0.f, s2 = 0.f, s3 = 0.f;
  for (int d = 0; d < 64; ++d) {
    float wg = Wgat[k * 256 + h * 64 + d];
    s1 += wg * att_src[h * 64 + d];
    s2 += wg * att_dst[h * 64 + d];
    s3 += Wge[k * 256 + h * 64 + d] * att_edge[h * 64 + d];
  }
  Asrc[k * 4 + h] = s1; Adst[k * 4 + h] = s2; WeA[k * 4 + h] = s3;
  for (int d = 0; d < 64; ++d)
    Wstack[(size_t)(k * 4 + h) * 64 + d] = 0.25f * Wgat[k * 256 + h * 64 + d];
}

__global__ void edge_post(const float* __restrict__ e, const int* __restrict__ dst,
                          const float* __restrict__ WeA, float* __restrict__ alphaE,
                          float* __restrict__ e_sum, float* __restrict__ deg)
{
  int ei = blockIdx.x * blockDim.x + threadIdx.x;
  if (ei >= NE) return;
  int d = dst[ei];
  const float* er = e + (size_t)ei * 64;
  float* es = e_sum + (size_t)d * 64;
  float a0 = 0.f, a1 = 0.f, a2 = 0.f, a3 = 0.f;
  for (int c = 0; c < 64; ++c) {
    float ec = er[c];
    a0 += ec * WeA[c * 4 + 0]; a1 += ec * WeA[c * 4 + 1];
    a2 += ec * WeA[c * 4 + 2]; a3 += ec * WeA[c * 4 + 3];
    atomicAdd(&es[c], ec);
  }
  alphaE[(size_t)ei * 4 + 0] = a0; alphaE[(size_t)ei * 4 + 1] = a1;
  alphaE[(size_t)ei * 4 + 2] = a2; alphaE[(size_t)ei * 4 + 3] = a3;
  atomicAdd(&deg[d], 1.0f);
}

__global__ void node_att(const float* __restrict__ h, const float* __restrict__ Asrc,
                         const float* __restrict__ Adst, float* asrc, float* adst)
{
  int n = blockIdx.x * blockDim.x + threadIdx.x;
  if (n >= NN) return;
  const float* hr = h + (size_t)n * 64;
  float s[4] = {0, 0, 0, 0}, d[4] = {0, 0, 0, 0};
  for (int c = 0; c < 64; ++c) {
    float hv = hr[c];
#pragma unroll
    for (int k = 0; k < 4; ++k) { s[k] += hv * Asrc[c * 4 + k]; d[k] += hv * Adst[c * 4 + k]; }
  }
#pragma unroll
  for (int k = 0; k < 4; ++k) { asrc[n * 4 + k] = s[k]; adst[n * 4 + k] = d[k]; }
}

__global__ void self_alpha(const float* __restrict__ e_sum, const float* __restrict__ deg,
                           const float* __restrict__ WeA, const float* __restrict__ asrc,
                           const float* __restrict__ adst, float* alphaS, unsigned* amax)
{
  int n = blockIdx.x * blockDim.x + threadIdx.x;
  if (n >= NN) return;
  float dg = fmaxf(deg[n], 1.0f);
  const float* es = e_sum + (size_t)n * 64;
  float a[4] = {0, 0, 0, 0};
  for (int c = 0; c < 64; ++c) {
    float em = es[c] / dg;
#pragma unroll
    for (int k = 0; k < 4; ++k) a[k] += em * WeA[c * 4 + k];
  }
#pragma unroll
  for (int k = 0; k < 4; ++k) {
    float al = asrc[n * 4 + k] + adst[n * 4 + k] + a[k];
    al = al > 0.f ? al : 0.2f * al;
    alphaS[n * 4 + k] = al;
    atomicMax(&amax[n * 4 + k], flipf(al));
  }
}

__global__ void edge_alpha(const int* __restrict__ src, const int* __restrict__ dst,
                           const float* __restrict__ asrc, const float* __restrict__ adst,
                           float* alphaE, unsigned* amax)
{
  int ei = blockIdx.x * blockDim.x + threadIdx.x;
  if (ei >= NE) return;
  int s = src[ei], d = dst[ei];
#pragma unroll
  for (int k = 0; k < 4; ++k) {
    float al = asrc[s * 4 + k] + adst[d * 4 + k] + alphaE[(size_t)ei * 4 + k];
    al = al > 0.f ? al : 0.2f * al;
    alphaE[(size_t)ei * 4 + k] = al;
    atomicMax(&amax[d * 4 + k], flipf(al));
  }
}

__global__ void self_exp(const float* __restrict__ alphaS, const unsigned* __restrict__ amax,
                         float* exS, float* denom)
{
  int n = blockIdx.x * blockDim.x + threadIdx.x;
  if (n >= NN) return;
#pragma unroll
  for (int k = 0; k < 4; ++k) {
    float ex = __expf(alphaS[n * 4 + k] - unflipf(amax[n * 4 + k]));
    exS[n * 4 + k] = ex;
    atomicAdd(&denom[n * 4 + k], ex);
  }
}

__global__ void edge_exp(const int* __restrict__ dst, const float* __restrict__ alphaE,
                         const unsigned* __restrict__ amax, float* exE, float* denom)
{
  int ei = blockIdx.x * blockDim.x + threadIdx.x;
  if (ei >= NE) return;
  int d = dst[ei];
#pragma unroll
  for (int k = 0; k < 4; ++k) {
    float ex = __expf(alphaE[(size_t)ei * 4 + k] - unflipf(amax[d * 4 + k]));
    exE[(size_t)ei * 4 + k] = ex;
    atomicAdd(&denom[d * 4 + k], ex);
  }
}

__global__ void self_agg(const float* __restrict__ h, const float* __restrict__ exS,
                         const float* __restrict__ denom, float* __restrict__ sbuf)
{
  int n = blockIdx.x * blockDim.x + threadIdx.x;
  if (n >= NN) return;
  float w[4];
#pragma unroll
  for (int k = 0; k < 4; ++k) w[k] = exS[n * 4 + k] / denom[n * 4 + k];
  const float* hr = h + (size_t)n * 64;
  float* sr = sbuf + (size_t)n * 256;
  for (int c = 0; c < 64; ++c) {
    float hv = hr[c];
#pragma unroll
    for (int k = 0; k < 4; ++k) sr[c * 4 + k] = w[k] * hv;
  }
}

__global__ void edge_agg(const int* __restrict__ src, const int* __restrict__ dst,
                         const float* __restrict__ h, const float* __restrict__ exE,
                         const float* __restrict__ denom, float* __restrict__ sbuf)
{
  int ei = blockIdx.x * blockDim.x + threadIdx.x;
  if (ei >= NE) return;
  int s = src[ei], d = dst[ei];
  float w[4];
#pragma unroll
  for (int k = 0; k < 4; ++k) w[k] = exE[(size_t)ei * 4 + k] / denom[d * 4 + k];
  const float* hr = h + (size_t)s * 64;
  float* sr = sbuf + (size_t)d * 256;
  for (int c = 0; c < 64; ++c) {
    float hv = hr[c];
#pragma unroll
    for (int k = 0; k < 4; ++k) atomicAdd(&sr[c * 4 + k], w[k] * hv);
  }
}

__global__ void gine_scatter(const int* __restrict__ src, const int* __restrict__ dst,
                             const float* __restrict__ hin, const float* __restrict__ e,
                             float* __restrict__ agg)
{
  int ei = blockIdx.x * blockDim.x + threadIdx.x;
  if (ei >= NE) return;
  int s = src[ei], d = dst[ei];
  const float* hr = hin + (size_t)s * 64;
  const float* er = e + (size_t)ei * 64;
  float* ar = agg + (size_t)d * 64;
  for (int c = 0; c < 64; ++c) atomicAdd(&ar[c], fmaxf(hr[c] + er[c], 0.0f));
}

__global__ void pool_kernel(const int* __restrict__ batch, const float* __restrict__ h1,
                            const float* __restrict__ h2, const float* __restrict__ h3,
                            float* __restrict__ pool)
{
  int n = blockIdx.x * blockDim.x + threadIdx.x;
  if (n >= NN) return;
  int g = batch[n];
  float* pr = pool + (size_t)g * 192;
  for (int c = 0; c < 64; ++c) {
    atomicAdd(&pr[c],       h1[(size_t)n * 64 + c]);
    atomicAdd(&pr[64 + c],  h2[(size_t)n * 64 + c]);
    atomicAdd(&pr[128 + c], h3[(size_t)n * 64 + c]);
  }
}

// ---------------- launcher ----------------
extern "C" void kernel_launch(void* const* d_in, const int* in_sizes, int n_in,
                              void* d_out, int out_size, void* d_ws, size_t ws_size,
                              hipStream_t stream)
{
  const float* x         = (const float*)d_in[0];
  const int*   src       = (const int*)d_in[1];
  const int*   dst       = src + NE;
  const float* edge_attr = (const float*)d_in[2];
  const int*   batch     = (const int*)d_in[3];
  const float *We1 = (const float*)d_in[4],  *be1 = (const float*)d_in[5];
  const float *We2 = (const float*)d_in[6],  *be2 = (const float*)d_in[7];
  const float *Wi1 = (const float*)d_in[8],  *bi1 = (const float*)d_in[9];
  const float *bnig = (const float*)d_in[10], *bnib = (const float*)d_in[11];
  const float *bnim = (const float*)d_in[12], *bniv = (const float*)d_in[13];
  const float *Wi2 = (const float*)d_in[14], *bi2 = (const float*)d_in[15];
  const float *Wgat = (const float*)d_in[16];
  const float *att_src = (const float*)d_in[17], *att_dst = (const float*)d_in[18];
  const float *Wge = (const float*)d_in[19], *att_edge = (const float*)d_in[20];
  const float *bgat = (const float*)d_in[21];
  const float *Wh1 = (const float*)d_in[22], *bh1 = (const float*)d_in[23];
  const float *bnhg = (const float*)d_in[24], *bnhb = (const float*)d_in[25];
  const float *bnhm = (const float*)d_in[26], *bnhv = (const float*)d_in[27];
  const float *Wh2 = (const float*)d_in[28], *bh2 = (const float*)d_in[29];
  const float *Wf1 = (const float*)d_in[30], *bf1 = (const float*)d_in[31];
  const float *Wf2 = (const float*)d_in[32], *bf2 = (const float*)d_in[33];

  size_t off = 0;
  float* base = (float*)d_ws;
  auto alloc = [&](size_t n) { float* p = base + off; off += n; return p; };
  float* f_e    = alloc((size_t)NE * 64);
  float* f_big  = alloc((size_t)NE * 64);   // et; later reused for sbuf/agg/pool/r1
  float* f_h    = alloc((size_t)NN * 64);
  float* f_tmp  = alloc((size_t)NN * 64);
  float* f_h1   = alloc((size_t)NN * 64);
  float* f_h2   = alloc((size_t)NN * 64);
  float* f_h3   = alloc((size_t)NN * 64);
  float* f_aE   = alloc((size_t)NE * 4);
  float* f_exE  = alloc((size_t)NE * 4);
  float* f_asrc = alloc((size_t)NN * 4);
  float* f_adst = alloc((size_t)NN * 4);
  float* f_aS   = alloc((size_t)NN * 4);
  float* f_exS  = alloc((size_t)NN * 4);
  float* f_den  = alloc((size_t)NN * 4);
  unsigned* f_amax = (unsigned*)alloc((size_t)NN * 4);
  float* f_deg  = alloc((size_t)NN);
  float* f_esum = alloc((size_t)NN * 64);
  float* f_Asrc = alloc(256);
  float* f_Adst = alloc(256);
  float* f_WeA  = alloc(256);
  float* f_Wst  = alloc(256 * 64);
  float* f_s    = f_big;                      // [NN,256]
  float* f_agg  = f_big + (size_t)NN * 256;   // [NN,64]
  float* f_pool = f_agg + (size_t)NN * 64;    // [NG,192]
  float* f_r1   = f_pool + (size_t)NG * 192;  // [NG,192]

  const int TB = 256;
  const int GB_E = (NE + TB - 1) / TB;
  const int GB_N = (NN + TB - 1) / TB;
  const int GM_N = (NN / 16 + 7) / 8;   // 8 m-tiles per 256-thread block
  const int GM_E = (NE / 16 + 7) / 8;
  const int GM_G = (NG / 16 + 7) / 8;
  auto zgrid = [](size_t n) { size_t b = (n + 255) / 256; return (int)(b < 4096 ? b : 4096); };
  auto shm = [](int K, int Nc) { return (size_t)K * Nc * sizeof(float); };

  // 0. fold attention vectors
  precompute<<<1, 256, 0, stream>>>(Wgat, att_src, att_dst, Wge, att_edge,
                                    f_Asrc, f_Adst, f_WeA, f_Wst);

  // 1. input MLP: x -> h                                [WMMA + TDM/LDS]
  gemm_wmma<4, false, true,  true,  true><<<GM_N, TB, shm(64, 64), stream>>>(
      x, nullptr, Wi1, bi1, bnig, bnib, bnim, bniv, f_tmp, NN, 64);
  gemm_wmma<4, false, false, true,  true><<<GM_N, TB, shm(64, 64), stream>>>(
      f_tmp, nullptr, Wi2, bi2, nullptr, nullptr, nullptr, nullptr, f_h, NN, 64);

  // 2. edge MLP: edge_attr -> e                         [WMMA + TDM/LDS]
  gemm_wmma<4, false, false, true,  true><<<GM_E, TB, shm(16, 64), stream>>>(
      edge_attr, nullptr, We1, be1, nullptr, nullptr, nullptr, nullptr, f_big, NE, 16);
  gemm_wmma<4, false, false, false, true><<<GM_E, TB, shm(64, 64), stream>>>(
      f_big, nullptr, We2, be2, nullptr, nullptr, nullptr, nullptr, f_e, NE, 64);

  // 3. GAT attention logits (attention vectors folded into [64,4] mats)
  node_att<<<GB_N, TB, 0, stream>>>(f_h, f_Asrc, f_Adst, f_asrc, f_adst);
  zero_f32<<<zgrid((size_t)NN * 77), 256, 0, stream>>>(
      f_den, (size_t)NN * 4 * 3 + NN + (size_t)NN * 64);  // denom, amax, deg, e_sum
  edge_post<<<GB_E, TB, 0, stream>>>(f_e, dst, f_WeA, f_aE, f_esum, f_deg);
  self_alpha<<<GB_N, TB, 0, stream>>>(f_esum, f_deg, f_WeA, f_asrc, f_adst, f_aS, f_amax);
  edge_alpha<<<GB_E, TB, 0, stream>>>(src, dst, f_asrc, f_adst, f_aE, f_amax);
  self_exp<<<GB_N, TB, 0, stream>>>(f_aS, f_amax, f_exS, f_den);
  edge_exp<<<GB_E, TB, 0, stream>>>(dst, f_aE, f_amax, f_exE, f_den);

  // 4. aggregate in h-space (64 floats/edge, not 256), project once [WMMA]
  zero_f32<<<zgrid((size_t)NN * 256), 256, 0, stream>>>(f_s, (size_t)NN * 256);
  self_agg<<<GB_N, TB, 0, stream>>>(f_h, f_exS, f_den, f_s);
  edge_agg<<<GB_E, TB, 0, stream>>>(src, dst, f_h, f_exE, f_den, f_s);
  gemm_wmma<4, false, false, false, true><<<GM_N, TB, shm(256, 64), stream>>>(
      f_s, nullptr, f_Wst, bgat, nullptr, nullptr, nullptr, nullptr, f_h1, NN, 256);

  // 5. GINE conv2: h1 -> h2                             [WMMA]
  zero_f32<<<zgrid((size_t)NN * 64), 256, 0, stream>>>(f_agg, (size_t)NN * 64);
  gine_scatter<<<GB_E, TB, 0, stream>>>(src, dst, f_h1, f_e, f_agg);
  gemm_wmma<4, true,  true,  true,  true><<<GM_N, TB, shm(64, 64), stream>>>(
      f_h1, f_agg, Wh1, bh1, bnhg, bnhb, bnhm, bnhv, f_tmp, NN, 64);
  gemm_wmma<4, false, false, true,  true><<<GM_N, TB, shm(64, 64), stream>>>(
      f_tmp, nullptr, Wh2, bh2, nullptr, nullptr, nullptr, nullptr, f_h2, NN, 64);

  // 6. GINE conv3: h2 -> h3                             [WMMA]
  zero_f32<<<zgrid((size_t)NN * 64), 256, 0, stream>>>(f_agg, (size_t)NN * 64);
  gine_scatter<<<GB_E, TB, 0, stream>>>(src, dst, f_h2, f_e, f_agg);
  gemm_wmma<4, true,  true,  true,  true><<<GM_N, TB, shm(64, 64), stream>>>(
      f_h2, f_agg, Wh1, bh1, bnhg, bnhb, bnhm, bnhv, f_tmp, NN, 64);
  gemm_wmma<4, false, false, true,  true><<<GM_N, TB, shm(64, 64), stream>>>(
      f_tmp, nullptr, Wh2, bh2, nullptr, nullptr, nullptr, nullptr, f_h3, NN, 64);

  // 7. pooling + readout (tiny; direct-global B)        [WMMA]
  zero_f32<<<zgrid((size_t)NG * 192), 256, 0, stream>>>(f_pool, (size_t)NG * 192);
  pool_kernel<<<GB_N, TB, 0, stream>>>(batch, f_h1, f_h2, f_h3, f_pool);
  gemm_wmma<12, false, false, true,  false><<<GM_G, TB, 0, stream>>>(
      f_pool, nullptr, Wf1, bf1, nullptr, nullptr, nullptr, nullptr, f_r1, NG, 192);
  gemm_wmma<2,  false, false, false, false><<<GM_G, TB, 0, stream>>>(
      f_r1, nullptr, Wf2, bf2, nullptr, nullptr, nullptr, nullptr, (float*)d_out, NG, 192);
}